// PhysicsInformedLoss_37598143709860
// MI455X (gfx1250) — compile-verified
//
#include <hip/hip_runtime.h>
#include <math.h>

typedef float v2f __attribute__((ext_vector_type(2)));
typedef float v8f __attribute__((ext_vector_type(8)));

#define K_PENALTY 10000.0f
#define EN_BLOCKS 128
#define EN_THREADS 256
#define CP_THREADS 1024

// ---------------------------------------------------------------------------
// Energy terms: pi_int + pi_ext summed over all (n,t); caller divides by T.
// ---------------------------------------------------------------------------
__global__ void energy_kernel(const float* __restrict__ disp,
                              const float* __restrict__ stress,
                              const float* __restrict__ strain,
                              const float* __restrict__ vol,
                              const float* __restrict__ bcv,
                              const int*   __restrict__ bct,
                              float* __restrict__ epart,
                              int N, int T) {
    __shared__ float sdata[EN_THREADS];
    const int NT = N * T;
    float acc = 0.f;
    for (int idx = blockIdx.x * blockDim.x + threadIdx.x; idx < NT;
         idx += gridDim.x * blockDim.x) {
        const float* st = stress + (size_t)idx * 6;
        const float* sn = strain + (size_t)idx * 6;
        float dc = st[0] * sn[0] + st[1] * sn[1] + st[2] * sn[2]
                 + 2.f * (st[3] * sn[3] + st[4] * sn[4] + st[5] * sn[5]);
        float ew = 0.f;
        if (bct[idx] == 2) {
            const float* b = bcv  + (size_t)idx * 3;
            const float* d = disp + (size_t)idx * 3;
            ew = b[0] * d[0] + b[1] * d[1] + b[2] * d[2];
        }
        acc += (0.5f * dc - ew) * vol[idx / T];
    }
    sdata[threadIdx.x] = acc;
    __syncthreads();
    for (int s = EN_THREADS / 2; s > 0; s >>= 1) {
        if ((int)threadIdx.x < s) sdata[threadIdx.x] += sdata[threadIdx.x + s];
        __syncthreads();
    }
    if (threadIdx.x == 0) epart[blockIdx.x] = sdata[0];
}

// ---------------------------------------------------------------------------
// Per-timestep compaction of contact nodes (bc_type==3).
// pos4 = (x, y, z, |p|^2); padded slots: pos=0, r=-1e30, dom=-1 -> zero terms.
// One block per timestep; deterministic block-scan ordering.
// ---------------------------------------------------------------------------
__global__ void compact_kernel(const float* __restrict__ coords,
                               const float* __restrict__ disp,
                               const float* __restrict__ vol,
                               const int*   __restrict__ bct,
                               const int*   __restrict__ dom,
                               float4* __restrict__ pos4,
                               float*  __restrict__ rad,
                               int*    __restrict__ domc,
                               int*    __restrict__ ncArr,
                               int N, int T, int NPAD) {
    const int t = blockIdx.x;
    float4* P = pos4 + (size_t)t * NPAD;
    float*  R = rad  + (size_t)t * NPAD;
    int*    D = domc + (size_t)t * NPAD;

    for (int i = threadIdx.x; i < NPAD; i += CP_THREADS) {
        P[i] = make_float4(0.f, 0.f, 0.f, 0.f);
        R[i] = -1.0e30f;
        D[i] = -1;
    }
    __syncthreads();

    __shared__ int scan[CP_THREADS];
    __shared__ int s_total;
    if (threadIdx.x == 0) s_total = 0;
    __syncthreads();

    for (int base = 0; base < N; base += CP_THREADS) {
        const int n = base + threadIdx.x;
        int flag = 0;
        if (n < N) flag = (bct[(size_t)n * T + t] == 3) ? 1 : 0;
        scan[threadIdx.x] = flag;
        __syncthreads();
        for (int off = 1; off < CP_THREADS; off <<= 1) {
            int v = 0;
            if ((int)threadIdx.x >= off) v = scan[threadIdx.x - off];
            __syncthreads();
            scan[threadIdx.x] += v;
            __syncthreads();
        }
        const int incl = scan[threadIdx.x];
        const int slot = s_total + incl - flag;   // exclusive position
        if (flag) {
            const size_t ci = (size_t)n * T + t;
            const float x = coords[ci * 3 + 0] + disp[ci * 3 + 0];
            const float y = coords[ci * 3 + 1] + disp[ci * 3 + 1];
            const float z = coords[ci * 3 + 2] + disp[ci * 3 + 2];
            P[slot] = make_float4(x, y, z, x * x + y * y + z * z);
            R[slot] = 0.5f * cbrtf(vol[n]);
            D[slot] = dom[ci];
        }
        __syncthreads();
        if (threadIdx.x == CP_THREADS - 1) s_total += incl;
        __syncthreads();
    }
    if (threadIdx.x == 0) ncArr[t] = s_total;
}

// ---------------------------------------------------------------------------
// Pairwise contact penalty. One wave (32 threads) per (i_tile, t); loops over
// j_tiles. Dot products via V_WMMA_F32_16X16X4_F32:
//   A: 16x4 (rows=i nodes, K=xyz0), lane l -> row l%16, K = 2*(l/16)+{0,1}
//   B: 4x16 (K=xyz0, cols=j nodes), same per-lane gather from the j tile
//   D: reg r, lane l -> (i_local = 8*(l/16)+r, j_local = l%16)
// Epilogue uses raw hardware V_SQRT_F32 (TRANS pipe, ~1 ulp) — no software
// refinement; co-executes with packed-FMA epilogue and WMMA issue.
// ---------------------------------------------------------------------------
__device__ __forceinline__ float tile_step(const float* __restrict__ P,
                                           const float* __restrict__ R,
                                           const int*   __restrict__ D,
                                           int jbase, int l16, int half,
                                           v2f a,
                                           const float* ri, const float* p2i,
                                           const int* di) {
    v2f b;
    const float* bp = P + (size_t)(jbase + l16) * 4 + 2 * half;
    b[0] = bp[0];
    b[1] = (half == 1) ? 0.f : bp[1];     // K=3 slot zeroed (K=xyz0)

    const int   j   = jbase + l16;
    const float rj  = R[j];
    const float p2j = P[(size_t)j * 4 + 3];
    const int   dj  = D[j];

    v8f c = {0.f, 0.f, 0.f, 0.f, 0.f, 0.f, 0.f, 0.f};
    c = __builtin_amdgcn_wmma_f32_16x16x4_f32(
            /*neg_a=*/false, a, /*neg_b=*/false, b,
            /*c_mod=*/(short)0, c, /*reuse_a=*/false, /*reuse_b=*/false);

    float acc = 0.f;
#pragma unroll
    for (int r = 0; r < 8; ++r) {
        const float sq   = p2i[r] + p2j - 2.f * c[r];
        const float s    = __builtin_amdgcn_sqrtf(sq);     // raw v_sqrt_f32
        const float dist = (sq > 0.f) ? s : 0.f;
        float pen = (ri[r] + rj) - dist;
        pen = (pen > 0.f) ? pen : 0.f;
        const float w = (di[r] != dj) ? 1.f : 0.f;
        acc += pen * pen * w;
    }
    return acc;
}

__global__ void contact_kernel(const float4* __restrict__ pos4,
                               const float*  __restrict__ rad,
                               const int*    __restrict__ domc,
                               const int*    __restrict__ ncArr,
                               float* __restrict__ cpart,
                               int NPAD, int ITILES) {
    const int itile = blockIdx.x;
    const int t     = blockIdx.y;
    const int lane  = threadIdx.x;      // 0..31
    const int half  = lane >> 4;        // 0 or 1
    const int l16   = lane & 15;
    const int nc    = ncArr[t];
    const int pidx  = t * ITILES + itile;

    if (itile * 16 >= nc || nc < 2) {   // uniform branch; EXEC stays full
        if (lane == 0) cpart[pidx] = 0.f;
        return;
    }

    const float* P = (const float*)(pos4 + (size_t)t * NPAD);
    const float* R = rad  + (size_t)t * NPAD;
    const int*   D = domc + (size_t)t * NPAD;

    const int ibase = itile * 16;

    // A operand (constant across the j loop)
    v2f a;
    {
        const float* ap = P + (size_t)(ibase + l16) * 4 + 2 * half;
        a[0] = ap[0];
        a[1] = (half == 1) ? 0.f : ap[1];
    }

    // i-tile scalars for the 8 rows this lane's accumulators cover
    float ri[8], p2i[8];
    int   di[8];
#pragma unroll
    for (int r = 0; r < 8; ++r) {
        const int i = ibase + 8 * half + r;
        ri[r]  = R[i];
        p2i[r] = P[(size_t)i * 4 + 3];
        di[r]  = D[i];
    }

    const int jtiles = (nc + 15) >> 4;
    float acc = 0.f;

    // Unrolled x2 so two tiles' loads/WMMAs/epilogues can overlap.
    int jt = 0;
    for (; jt + 2 <= jtiles; jt += 2) {
        acc += tile_step(P, R, D, (jt    ) * 16, l16, half, a, ri, p2i, di);
        acc += tile_step(P, R, D, (jt + 1) * 16, l16, half, a, ri, p2i, di);
    }
    for (; jt < jtiles; ++jt)
        acc += tile_step(P, R, D, jt * 16, l16, half, a, ri, p2i, di);

    // wave32 reduction
    for (int off = 16; off > 0; off >>= 1)
        acc += __shfl_xor(acc, off, 32);
    if (lane == 0) cpart[pidx] = acc;
}

// ---------------------------------------------------------------------------
// Deterministic serial combine (work is ~2K adds).
// ---------------------------------------------------------------------------
__global__ void finalize_kernel(const float* __restrict__ epart,
                                const float* __restrict__ cpart,
                                const int*   __restrict__ ncArr,
                                float* __restrict__ out,
                                int T, int ITILES, float invT) {
    if (threadIdx.x != 0 || blockIdx.x != 0) return;
    float E = 0.f;
    for (int i = 0; i < EN_BLOCKS; ++i) E += epart[i];
    E *= invT;
    float s = 0.f, cnt = 0.f;
    for (int t = 0; t < T; ++t) {
        float cs = 0.f;
        for (int i = 0; i < ITILES; ++i) cs += cpart[t * ITILES + i];
        if (ncArr[t] >= 2) { s += K_PENALTY * cs; cnt += 1.f; }
    }
    const float lc = (cnt > 0.f) ? (s / cnt) : 0.f;
    out[0] = E + lc;
}

// ---------------------------------------------------------------------------
extern "C" void kernel_launch(void* const* d_in, const int* in_sizes, int n_in,
                              void* d_out, int out_size, void* d_ws, size_t ws_size,
                              hipStream_t stream) {
    const float* disp   = (const float*)d_in[0];
    const float* stress = (const float*)d_in[1];
    const float* strain = (const float*)d_in[2];
    const float* coords = (const float*)d_in[3];
    const float* vol    = (const float*)d_in[4];
    const float* bcv    = (const float*)d_in[5];
    const int*   bct    = (const int*)d_in[6];
    const int*   dom    = (const int*)d_in[7];

    const int N    = in_sizes[4];            // volume is [N,1]
    const int T    = in_sizes[6] / N;        // bc_type is [N,T]
    const int NPAD = ((N + 15) / 16) * 16;
    const int ITILES = NPAD / 16;

    // Workspace carve-up (d_ws is 256B-aligned; float4 array first).
    char* ws = (char*)d_ws;
    size_t off = 0;
    float4* pos4 = (float4*)(ws + off); off += (size_t)T * NPAD * sizeof(float4);
    float*  rad  = (float*)(ws + off);  off += (size_t)T * NPAD * sizeof(float);
    int*    domc = (int*)(ws + off);    off += (size_t)T * NPAD * sizeof(int);
    float*  epart = (float*)(ws + off); off += (size_t)EN_BLOCKS * sizeof(float);
    float*  cpart = (float*)(ws + off); off += (size_t)T * ITILES * sizeof(float);
    int*    ncArr = (int*)(ws + off);   off += (size_t)T * sizeof(int);
    (void)ws_size; (void)n_in; (void)out_size;

    energy_kernel<<<EN_BLOCKS, EN_THREADS, 0, stream>>>(
        disp, stress, strain, vol, bcv, bct, epart, N, T);

    compact_kernel<<<T, CP_THREADS, 0, stream>>>(
        coords, disp, vol, bct, dom, pos4, rad, domc, ncArr, N, T, NPAD);

    contact_kernel<<<dim3(ITILES, T), 32, 0, stream>>>(
        pos4, rad, domc, ncArr, cpart, NPAD, ITILES);

    finalize_kernel<<<1, 1, 0, stream>>>(
        epart, cpart, ncArr, (float*)d_out, T, ITILES, 1.0f / (float)T);
}